// PINN_26121991095018
// MI455X (gfx1250) — compile-verified
//
#include <hip/hip_runtime.h>

// CDNA5 / gfx1250: wave32, WMMA 16x16 shapes. f32 WMMA (16x16x4) used for
// full-precision PINN second derivatives. D = A(16x4) x B(4x16) + C(16x16).
typedef __attribute__((ext_vector_type(2))) float v2f;
typedef __attribute__((ext_vector_type(8))) float v8f;

#define W_DIM      64
#define H_STRIDE   68   // padded H row stride (floats): conflict-free A b64 loads
#define W_STRIDE   72   // padded W row stride (floats): conflict-free B loads
#define WAVES_PB   2
#define THREADS_PB (WAVES_PB * 32)
#define PTS_PB     (WAVES_PB * 16)
#define NU_F       0.003183098861837907f   // 0.01 / pi

struct PinnParams {
  const float* t;
  const float* x;
  const float* W[8];
  const float* b[8];
  const float* lb;
  const float* ub;
  float* out;
  int n;
};

__global__ __launch_bounds__(THREADS_PB)
void pinn_burgers_wmma(PinnParams p) {
  // Per-block: one layer's weights (staged per layer), biases, and per-wave
  // H buffers holding 4 forward-mode streams (h, h_t, h_x, h_xx), 16 pts x 64.
  __shared__ __align__(16) float sW[W_DIM * W_STRIDE];
  __shared__ float sB[W_DIM];
  __shared__ __align__(16) float sH[WAVES_PB][4][16 * H_STRIDE];

  const int tid  = threadIdx.x;
  const int wave = tid >> 5;
  const int lane = tid & 31;
  const int half = lane >> 4;   // which 16-lane half of the wave
  const int lm   = lane & 15;
  const int base = blockIdx.x * PTS_PB + wave * 16;

  // ---------------- Layer 0: input normalize + (2 -> 64) + tanh combine ----
  const float lb0 = p.lb[0], lb1 = p.lb[1];
  const float ub0 = p.ub[0], ub1 = p.ub[1];
  const float st = 2.0f / (ub0 - lb0);   // dh0/dt
  const float sx = 2.0f / (ub1 - lb1);   // dh1/dx
  {
    const float* W0 = p.W[0];
    const float* b0 = p.b[0];
    for (int idx = lane; idx < 16 * W_DIM; idx += 32) {
      int m = idx >> 6;
      int j = idx & (W_DIM - 1);
      float h0  = (p.t[base + m] - lb0) * st - 1.0f;
      float h1  = (p.x[base + m] - lb1) * sx - 1.0f;
      float w0j = W0[j], w1j = W0[W_DIM + j];
      float a   = fmaf(h0, w0j, fmaf(h1, w1j, b0[j]));
      float at  = st * w0j;           // d a / dt
      float ax  = sx * w1j;           // d a / dx   (a_xx == 0 at layer 0)
      float g   = tanhf(a);
      float gp  = 1.0f - g * g;
      int o = m * H_STRIDE + j;
      sH[wave][0][o] = g;
      sH[wave][1][o] = gp * at;
      sH[wave][2][o] = gp * ax;
      sH[wave][3][o] = -2.0f * g * gp * ax * ax;
    }
  }

  // ---------------- Hidden layers 1..6: 4-stream GEMM on the WMMA pipe -----
  for (int L = 1; L <= 6; ++L) {
    __syncthreads();                       // previous sW fully consumed
    const float* Wl = p.W[L];
    for (int i4 = tid * 4; i4 < W_DIM * W_DIM; i4 += THREADS_PB * 4) {
      int row = i4 >> 6, col = i4 & (W_DIM - 1);
      *(float4*)&sW[row * W_STRIDE + col] = *(const float4*)&Wl[i4];
    }
    if (tid < W_DIM) sB[tid] = p.b[L][tid];
    __syncthreads();

    // Preload ALL A fragments (4 streams x K=64) into registers so the H
    // buffer can be overwritten in place as D tiles are produced.
    // ISA A layout (16x4 f32): lane = M (mod 16), K = vgpr + 2*half.
    v2f Ah[16], At[16], Ax[16], Aq[16];
    const int arow = lm * H_STRIDE + 2 * half;
#pragma unroll
    for (int k = 0; k < 16; ++k) {
      int o = arow + 4 * k;
      Ah[k] = *(const v2f*)&sH[wave][0][o];
      At[k] = *(const v2f*)&sH[wave][1][o];
      Ax[k] = *(const v2f*)&sH[wave][2][o];
      Aq[k] = *(const v2f*)&sH[wave][3][o];
    }

    for (int tau = 0; tau < 4; ++tau) {    // 4 N-tiles of 16 output features
      const int wcol = tau * 16 + lm;      // this lane's N column
      float bias = sB[wcol];
      v8f ch = {bias, bias, bias, bias, bias, bias, bias, bias};
      v8f ct = {}; v8f cx = {}; v8f cq = {};
#pragma unroll
      for (int k = 0; k < 16; ++k) {       // K = 64 in steps of 4
        int wrow = 4 * k + 2 * half;       // B layout: K = elem + 2*half, N = lane
        v2f Bf = { sW[wrow * W_STRIDE + wcol],
                   sW[(wrow + 1) * W_STRIDE + wcol] };
        ch = __builtin_amdgcn_wmma_f32_16x16x4_f32(false, Ah[k], false, Bf, (short)0, ch, false, false);
        ct = __builtin_amdgcn_wmma_f32_16x16x4_f32(false, At[k], false, Bf, (short)0, ct, false, false);
        cx = __builtin_amdgcn_wmma_f32_16x16x4_f32(false, Ax[k], false, Bf, (short)0, cx, false, false);
        cq = __builtin_amdgcn_wmma_f32_16x16x4_f32(false, Aq[k], false, Bf, (short)0, cq, false, false);
      }
      // tanh combine on the C/D layout: M = vgpr + 8*half, N = lane.
#pragma unroll
      for (int r = 0; r < 8; ++r) {
        float g   = tanhf(ch[r]);
        float gp  = 1.0f - g * g;
        float atv = ct[r], axv = cx[r], aqv = cq[r];
        int o = (r + 8 * half) * H_STRIDE + wcol;
        sH[wave][0][o] = g;
        sH[wave][1][o] = gp * atv;
        sH[wave][2][o] = gp * axv;
        sH[wave][3][o] = fmaf(gp, aqv, -2.0f * g * gp * axv * axv);
      }
    }
  }

  // ---------------- Layer 7 (64 -> 1) + Burgers residual -------------------
  {
    const float* W7 = p.W[7];
    float su = 0.0f, sut = 0.0f, sux = 0.0f, suq = 0.0f;
    const int hrow = lm * H_STRIDE;
#pragma unroll 8
    for (int j = 0; j < 32; ++j) {
      int jj = half * 32 + j;              // each half-wave sums half the features
      float w = W7[jj];
      su  = fmaf(sH[wave][0][hrow + jj], w, su);
      sut = fmaf(sH[wave][1][hrow + jj], w, sut);
      sux = fmaf(sH[wave][2][hrow + jj], w, sux);
      suq = fmaf(sH[wave][3][hrow + jj], w, suq);
    }
    su  += __shfl_down(su, 16);
    sut += __shfl_down(sut, 16);
    sux += __shfl_down(sux, 16);
    suq += __shfl_down(suq, 16);
    if (half == 0) {
      float u = su + p.b[7][0];
      float f = fmaf(u, sux, sut) - NU_F * suq;
      p.out[base + lm]       = u;
      p.out[p.n + base + lm] = f;
    }
  }
}

extern "C" void kernel_launch(void* const* d_in, const int* in_sizes, int n_in,
                              void* d_out, int out_size, void* d_ws, size_t ws_size,
                              hipStream_t stream) {
  // dict order: t, x, Ws[0..7], bs[0..7], lb, ub  -> 20 inputs
  PinnParams p;
  p.t = (const float*)d_in[0];
  p.x = (const float*)d_in[1];
  for (int i = 0; i < 8; ++i) p.W[i] = (const float*)d_in[2 + i];
  for (int i = 0; i < 8; ++i) p.b[i] = (const float*)d_in[10 + i];
  p.lb  = (const float*)d_in[18];
  p.ub  = (const float*)d_in[19];
  p.out = (float*)d_out;
  p.n   = in_sizes[0];               // 262144
  int blocks = p.n / PTS_PB;         // 8192 blocks of 64 threads (2 waves)
  pinn_burgers_wmma<<<blocks, THREADS_PB, 0, stream>>>(p);
}